// MedianFilter2D_68745246540291
// MI455X (gfx1250) — compile-verified
//
#include <hip/hip_runtime.h>
#include <stdint.h>

// 3x3 median filter, reflect padding, fp32, 48 planes of 512x512.
// Memory-bound (~100 MB traffic -> ~4.3us at 23.3 TB/s HBM). Strategy:
//  - async global->LDS staging (CDNA5 GLOBAL_LOAD_ASYNC_TO_LDS_B32, ASYNCcnt)
//  - non-temporal output stores (write-once data; keep L2 for input halo reuse)
//  - pruned 19-exchange median-of-9 network (30 v_min/max_f32 per pixel)
//  - 64x16 output tile per 256-thread block, 4 rows per thread with
//    row-sliding reuse (9 LDS loads first window, 3 per subsequent row).

#define IMG_H 512
#define IMG_W 512
#define TW 64
#define TH 16
#define HALO_W (TW + 2)          // 66
#define HALO_H (TH + 2)          // 18
#define HALO_N (HALO_W * HALO_H) // 1188
#define NTHREADS 256

__device__ __forceinline__ int reflect_idx(int i, int n) {
    // K=3 => only +/-1 can go out of range
    i = (i < 0) ? -i : i;
    i = (i >= n) ? (2 * n - 2 - i) : i;
    return i;
}

// One per-lane async B32 copy: global -> LDS, tracked by ASYNCcnt.
__device__ __forceinline__ void async_copy_b32(uint32_t lds_byte_off, const float* gptr) {
    asm volatile("global_load_async_to_lds_b32 %0, %1, off"
                 :: "v"(lds_byte_off), "v"(gptr)
                 : "memory");
}

__device__ __forceinline__ void wait_asynccnt0() {
    asm volatile("s_wait_asynccnt 0x0" ::: "memory");
}

// Pruned Devillard/Smith 19-exchange median-of-9 (30 min/max ops).
__device__ __forceinline__ float median9(float p0, float p1, float p2,
                                         float p3, float p4, float p5,
                                         float p6, float p7, float p8) {
#define S2(a, b) { float _t = fminf(a, b); b = fmaxf(a, b); a = _t; }
    S2(p1, p2); S2(p4, p5); S2(p7, p8);
    S2(p0, p1); S2(p3, p4); S2(p6, p7);
    S2(p1, p2); S2(p4, p5); S2(p7, p8);
    p3 = fmaxf(p0, p3);          // (0,3) max side only
    p5 = fminf(p5, p8);          // (5,8) min side only
    S2(p4, p7);                  // (4,7) both sides live
    p6 = fmaxf(p3, p6);          // (3,6) max side only
    p4 = fmaxf(p1, p4);          // (1,4) max side only
    p2 = fminf(p2, p5);          // (2,5) min side only
    p4 = fminf(p4, p7);          // (4,7) min side only
    float mn = fminf(p4, p2);    // (4,2) both sides, from pre-update values
    float mx = fmaxf(p4, p2);
    p4 = fmaxf(p6, mn);          // (6,4) max side only
    return fminf(p4, mx);        // (4,2) min side only
#undef S2
}

__global__ __launch_bounds__(NTHREADS)
void median3x3_async_kernel(const float* __restrict__ in, float* __restrict__ out) {
    __shared__ float tile[HALO_N];

    const int plane = blockIdx.z;
    const int gx0 = blockIdx.x * TW;
    const int gy0 = blockIdx.y * TH;
    const float* src = in + (size_t)plane * (IMG_H * IMG_W);

    // ---- Stage halo tile into LDS with async copies (5 rounds of 256 lanes)
    for (int i = threadIdx.x; i < HALO_N; i += NTHREADS) {
        const int r = i / HALO_W;
        const int c = i - r * HALO_W;
        const int gy = reflect_idx(gy0 + r - 1, IMG_H);
        const int gx = reflect_idx(gx0 + c - 1, IMG_W);
        const uint32_t lds_off = (uint32_t)(uintptr_t)(&tile[i]); // low 32b = LDS byte addr
        async_copy_b32(lds_off, src + (size_t)gy * IMG_W + gx);
    }
    wait_asynccnt0();   // this wave's async transfers complete
    __syncthreads();    // all waves' transfers visible in LDS

    // ---- Compute: thread (tx,ty) produces rows ty*4 .. ty*4+3 at column tx
    const int tx = threadIdx.x & (TW - 1);
    const int ty = threadIdx.x >> 6;        // 0..3
    const int base = ty * 4;                // first local output row

    const float* trow = tile + base * HALO_W + tx;
    float p0 = trow[0], p1 = trow[1], p2 = trow[2];
    trow += HALO_W;
    float p3 = trow[0], p4 = trow[1], p5 = trow[2];

    float* dst = out + (size_t)plane * (IMG_H * IMG_W)
                     + (size_t)(gy0 + base) * IMG_W + (gx0 + tx);

#pragma unroll
    for (int sub = 0; sub < 4; ++sub) {
        trow += HALO_W;
        const float p6 = trow[0], p7 = trow[1], p8 = trow[2];
        const float m = median9(p0, p1, p2, p3, p4, p5, p6, p7, p8);
        __builtin_nontemporal_store(m, dst);   // write-once output: TH_STORE_NT
        dst += IMG_W;
        // slide window down one row
        p0 = p3; p1 = p4; p2 = p5;
        p3 = p6; p4 = p7; p5 = p8;
    }
}

extern "C" void kernel_launch(void* const* d_in, const int* in_sizes, int n_in,
                              void* d_out, int out_size, void* d_ws, size_t ws_size,
                              hipStream_t stream) {
    (void)n_in; (void)d_ws; (void)ws_size; (void)out_size;
    const float* img = (const float*)d_in[0];
    float* out = (float*)d_out;
    const int planes = in_sizes[0] / (IMG_H * IMG_W);   // 16*3 = 48
    dim3 grid(IMG_W / TW, IMG_H / TH, planes);
    median3x3_async_kernel<<<grid, NTHREADS, 0, stream>>>(img, out);
}